// GridEncoder_36515811951009
// MI455X (gfx1250) — compile-verified
//
#include <hip/hip_runtime.h>
#include <hip/hip_bf16.h>
#include <math.h>

// ---------------- problem constants (from reference) ----------------
#define NUM_LEVELS 16
#define LOG2_T 19
#define TSZ (1u << LOG2_T)          // 524288 entries per level
#define PRIME1 2654435761u
#define PRIME2 805459861u
#define L0_ENTRIES 4096u            // 16^3 dense entries at level 0 (BASE_RES=16)

typedef float    v2f __attribute__((ext_vector_type(2)));
typedef unsigned v4u __attribute__((ext_vector_type(4)));
typedef int      v8i __attribute__((ext_vector_type(8)));
typedef int      v4i __attribute__((ext_vector_type(4)));

struct LevelParams {
    float    scale[NUM_LEVELS];
    unsigned res[NUM_LEVELS];
    unsigned dense_mask;            // bit l set => dense indexing at level l
};

__global__ __launch_bounds__(256) void grid_encode_kernel(
    const float* __restrict__ x,     // [N,3]
    const float* __restrict__ emb,   // [16, T, 2]
    float* __restrict__ out,         // [N, 32]
    unsigned n_points,
    LevelParams p)
{
    // Level-0 dense table staged in LDS (32 KB). 8 blocks/WGP -> 256 KB of 320 KB.
    __shared__ v2f lds0[L0_ENTRIES];

    const v2f* emb2 = (const v2f*)emb;

#if defined(__has_builtin) && __has_builtin(__builtin_amdgcn_tensor_load_to_lds)
    // --- CDNA5 Tensor Data Mover: DMA the 32 KB level-0 table into LDS ---
    if (threadIdx.x < 32) {          // wave-uniform: only wave 0 issues the TDM op
        unsigned long long ga = (unsigned long long)(const void*)emb;   // level 0 @ offset 0
        unsigned lb = (unsigned)(unsigned long long)(const void*)&lds0[0];
        v4u g0;
        g0[0] = 1u;                                          // count=1, user mode
        g0[1] = lb;                                          // lds_addr
        g0[2] = (unsigned)(ga & 0xffffffffu);                // global_addr[31:0]
        g0[3] = (unsigned)((ga >> 32) & 0x1ffffffu) | (2u << 30);  // addr[56:32] | type=2
        const unsigned E = L0_ENTRIES;                       // 8-byte elements
        v8i g1;
        g1[0] = (int)(3u << 16);                             // data_size=3 (8B); no mask/pad
        g1[1] = (int)((E & 0xffffu) << 16);                  // tensor_dim0[15:0]
        g1[2] = (int)((E >> 16) | (1u << 16));               // tensor_dim0[31:16] | tensor_dim1=1
        g1[3] = (int)(E << 16);                              // tile_dim0 = E
        g1[4] = 1;                                           // tile_dim1=1, tile_dim2=0
        g1[5] = (int)E;                                      // tensor_dim0_stride
        g1[6] = 0; g1[7] = 0;
        v4i g2 = {0, 0, 0, 0};
        v4i g3 = {0, 0, 0, 0};
        v8i g4 = {0, 0, 0, 0, 0, 0, 0, 0};                   // extra group (clang-23 6-arg form)
        __builtin_amdgcn_tensor_load_to_lds(g0, g1, g2, g3, g4, 0);
        __builtin_amdgcn_s_wait_tensorcnt(0);
    }
#else
    for (unsigned k = threadIdx.x; k < L0_ENTRIES; k += blockDim.x)
        lds0[k] = emb2[k];
#endif
    __syncthreads();

    const unsigned total  = n_points * NUM_LEVELS;           // one thread per (point, level)
    const unsigned stride = gridDim.x * blockDim.x;
    const unsigned tmask  = TSZ - 1u;
    const unsigned tid0   = blockIdx.x * blockDim.x + threadIdx.x;

    // l is invariant across the grid-stride loop (stride is a multiple of 16)
    const unsigned l     = tid0 & 15u;
    const float    s     = p.scale[l];
    const unsigned res   = p.res[l];
    const unsigned dense = (p.dense_mask >> l) & 1u;
    const unsigned rm1   = res - 1u;
    const unsigned rr    = res * res;
    const v2f* lvlbase   = emb2 + (size_t)l * TSZ;
    // generic pointer: LDS for level 0, global for others -> per-corner flat_load_b64
    const v2f* gbase     = (l == 0u) ? (const v2f*)&lds0[0] : lvlbase;

    for (unsigned i = tid0; i < total; i += stride) {
        const unsigned n  = i >> 4;
        const float*   xp = x + 3u * n;
        // prefetch next iteration's coordinates (global_prefetch_b8)
        unsigned pn = (i + stride) >> 4;
        pn = pn < n_points ? pn : (n_points - 1u);
        __builtin_prefetch(x + 3u * pn, 0, 1);

        const float px = xp[0], py = xp[1], pz = xp[2];
        const float fx = fmaf(px, s, 0.5f);
        const float fy = fmaf(py, s, 0.5f);
        const float fz = fmaf(pz, s, 0.5f);
        const float gx = floorf(fx), gy = floorf(fy), gz = floorf(fz);
        const float wx1 = fx - gx, wy1 = fy - gy, wz1 = fz - gz;
        const float wx0 = 1.0f - wx1, wy0 = 1.0f - wy1, wz0 = 1.0f - wz1;
        const unsigned bx = (unsigned)gx, by = (unsigned)gy, bz = (unsigned)gz;

        const unsigned x0 = min(bx, rm1),      x1 = min(bx + 1u, rm1);
        const unsigned y0 = min(by, rm1),      y1 = min(by + 1u, rm1);
        const unsigned z0 = min(bz, rm1),      z1 = min(bz + 1u, rm1);

        // per-axis contribution, selected branchlessly between dense and hashed
        const unsigned ty0 = dense ? y0 * res : y0 * PRIME1;
        const unsigned ty1 = dense ? y1 * res : y1 * PRIME1;
        const unsigned tz0 = dense ? z0 * rr  : z0 * PRIME2;
        const unsigned tz1 = dense ? z1 * rr  : z1 * PRIME2;

        #define IDX(xc, ty, tz) (dense ? ((xc) + (ty) + (tz)) \
                                       : ((((xc) ^ (ty)) ^ (tz)) & tmask))
        const unsigned i000 = IDX(x0, ty0, tz0);
        const unsigned i100 = IDX(x1, ty0, tz0);
        const unsigned i010 = IDX(x0, ty1, tz0);
        const unsigned i110 = IDX(x1, ty1, tz0);
        const unsigned i001 = IDX(x0, ty0, tz1);
        const unsigned i101 = IDX(x1, ty0, tz1);
        const unsigned i011 = IDX(x0, ty1, tz1);
        const unsigned i111 = IDX(x1, ty1, tz1);
        #undef IDX

        // 8 independent gathers (flat_load_b64) issued back-to-back
        const v2f v000 = gbase[i000];
        const v2f v100 = gbase[i100];
        const v2f v010 = gbase[i010];
        const v2f v110 = gbase[i110];
        const v2f v001 = gbase[i001];
        const v2f v101 = gbase[i101];
        const v2f v011 = gbase[i011];
        const v2f v111 = gbase[i111];

        const float wyz00 = wy0 * wz0, wyz10 = wy1 * wz0;
        const float wyz01 = wy0 * wz1, wyz11 = wy1 * wz1;

        v2f acc = v000 * (wx0 * wyz00);
        acc += v100 * (wx1 * wyz00);
        acc += v010 * (wx0 * wyz10);
        acc += v110 * (wx1 * wyz10);
        acc += v001 * (wx0 * wyz01);
        acc += v101 * (wx1 * wyz01);
        acc += v011 * (wx0 * wyz11);
        acc += v111 * (wx1 * wyz11);

        // streaming output (256 MB): non-temporal so the 64 MB of tables stay in L2
        __builtin_nontemporal_store(acc, (v2f*)out + i);
    }
}

extern "C" void kernel_launch(void* const* d_in, const int* in_sizes, int n_in,
                              void* d_out, int out_size, void* d_ws, size_t ws_size,
                              hipStream_t stream) {
    const float* x   = (const float*)d_in[0];
    const float* emb = (const float*)d_in[1];
    float*       out = (float*)d_out;
    const unsigned n_points = (unsigned)(in_sizes[0] / 3);

    // Level params computed exactly like the reference (double precision).
    LevelParams p;
    const double b = exp2(log2(2048.0 / 16.0) / (double)(NUM_LEVELS - 1));
    unsigned dm = 0;
    for (int l = 0; l < NUM_LEVELS; ++l) {
        const double sc = 16.0 * pow(b, (double)l) - 1.0;
        const long long r = (long long)ceil(sc) + 1;
        p.scale[l] = (float)sc;
        p.res[l]   = (unsigned)r;
        if (r * r * r <= (long long)TSZ) dm |= (1u << l);
    }
    p.dense_mask = dm;

    // Persistent grid: 2048 blocks x 256 threads (one TDM staging per block),
    // grid-stride over n_points*16 (point, level) units.
    dim3 grid(2048), block(256);
    grid_encode_kernel<<<grid, block, 0, stream>>>(x, emb, out, n_points, p);
}